// DeformConv_8177617732002
// MI455X (gfx1250) — compile-verified
//
#include <hip/hip_runtime.h>
#include <hip/hip_bf16.h>
#include <math.h>

// Problem constants (match reference)
#define BB   8
#define CC   64
#define CO_  64
#define HH   128
#define WW   128
#define K2C  9
#define NOFF 27            // 3*K2
#define HW   (HH*WW)       // 16384
#define BN_EPS 1e-5f

typedef __attribute__((ext_vector_type(2))) float v2f;
typedef __attribute__((ext_vector_type(8))) float v8f;

// ---------------------------------------------------------------------------
// Kernel 1: 3x3 offset conv as implicit GEMM on the WMMA f32 pipe.
//   M = 16 spatial positions per wave (one row segment)
//   N = 32 (27 offset channels padded to 2 tiles of 16, pad cols weight=0)
//   K = C*K2 = 576, stepped by 4 (V_WMMA_F32_16X16X4_F32)
// A element = zero-padded x[b, c, h+ky, w+kx]  (1 load + select)
// B fragments pre-permuted into 72 KB dynamic LDS.
// Output: raw off[b][27][h][w] planes in workspace.
// ---------------------------------------------------------------------------
__global__ void offset_kernel(const float* __restrict__ xin,
                              const float* __restrict__ w_off,
                              const float* __restrict__ b_off,
                              float* __restrict__ offA)
{
    extern __shared__ float ldsw[];                 // 18432 floats = 72 KB
    const int total = K2C * 16 * 2 * 32 * 2;        // [k][c4][nt][lane][2]
    for (int i = threadIdx.x; i < total; i += blockDim.x) {
        const int p    = i & 1;
        const int lane = (i >> 1) & 31;
        const int nt   = (i >> 6) & 1;
        const int c4   = (i >> 7) & 15;
        const int k    = (i >> 11);
        const int c    = c4 * 4 + 2 * (lane >> 4) + p;
        const int j    = nt * 16 + (lane & 15);     // offset-conv out channel
        ldsw[i] = (j < NOFF) ? w_off[(j * CC + c) * K2C + k] : 0.0f;
    }
    __syncthreads();

    const int lane   = threadIdx.x & 31;
    const int waveId = threadIdx.x >> 5;
    const int tile   = blockIdx.x * 8 + waveId;      // 8192 tiles total
    const int b      = tile / (HW / 16);
    const int rem    = tile % (HW / 16);
    const int h      = rem / (WW / 16);
    const int w0     = (rem % (WW / 16)) * 16;
    const int pos    = lane & 15;
    const int wx     = w0 + pos;
    const int kpair  = lane >> 4;

    v8f acc0 = {}, acc1 = {};
    const float* xb = xin + (size_t)b * CC * HW;

    for (int k = 0; k < K2C; ++k) {
        const int ky = k / 3 - 1, kx = k % 3 - 1;
        const int yy = h + ky, cl = wx + kx;
        const float vflag = (((yy >= 0) & (yy < HH) & (cl >= 0) & (cl < WW))) ? 1.f : 0.f;
        const int iofs = min(max(yy, 0), HH - 1) * WW + min(max(cl, 0), WW - 1);

        const float* pc  = xb + (2 * kpair) * HW;
        const v2f*   ldk = (const v2f*)(ldsw + k * (16 * 2 * 32 * 2));

        #pragma unroll 4
        for (int c4 = 0; c4 < 16; ++c4) {
            const float* p0 = pc + c4 * 4 * HW;
            const float a0 = vflag * p0[iofs];
            const float a1 = vflag * p0[HW + iofs];
            v2f A; A.x = a0; A.y = a1;

            const v2f* bp = ldk + c4 * (2 * 32);
            const v2f B0 = bp[lane];
            const v2f B1 = bp[32 + lane];

            acc0 = __builtin_amdgcn_wmma_f32_16x16x4_f32(false, A, false, B0, (short)0, acc0, false, false);
            acc1 = __builtin_amdgcn_wmma_f32_16x16x4_f32(false, A, false, B1, (short)0, acc1, false, false);
        }
    }

    // Epilogue (after all WMMAs; divergence here is safe)
    v8f accs[2] = { acc0, acc1 };
    const int mbase = 8 * (lane >> 4);
    #pragma unroll
    for (int nt = 0; nt < 2; ++nt) {
        const int j = nt * 16 + (lane & 15);
        if (j < NOFF) {
            const float bias = b_off[j];
            float* po = offA + ((size_t)b * NOFF + j) * HW + h * WW + w0;
            #pragma unroll
            for (int r = 0; r < 8; ++r)
                po[mbase + r] = accs[nt][r] + bias;
        }
    }
}

// ---------------------------------------------------------------------------
// Kernel 2: deformable gather + implicit GEMM via V_WMMA_F32_16X16X4_F32
//           + fused bias/BN/ReLU epilogue.
// One wave = 16 spatial positions x CO=64 (4 accumulator tiles).
// dy[k]=off[2k], dx[k]=off[2k+1], m[k]=sigmoid(off[18+k]) read from workspace.
// Main weights pre-permuted into 144 KB dynamic LDS in B-fragment order.
// ---------------------------------------------------------------------------
__global__ void deform_gemm_kernel(const float* __restrict__ xin,
                                   const float* __restrict__ wmain,
                                   const float* __restrict__ bconv,
                                   const float* __restrict__ gamma,
                                   const float* __restrict__ beta,
                                   const float* __restrict__ rmean,
                                   const float* __restrict__ rvar,
                                   const float* __restrict__ offA,
                                   float* __restrict__ out)
{
    extern __shared__ float ldsw[];                 // 36864 floats = 144 KB
    const int total = K2C * 16 * 4 * 32 * 2;        // [k][c4][nt][lane][2]
    for (int i = threadIdx.x; i < total; i += blockDim.x) {
        const int p    = i & 1;
        const int lane = (i >> 1) & 31;
        const int nt   = (i >> 6) & 3;
        const int c4   = (i >> 8) & 15;
        const int k    = (i >> 12);
        const int c    = c4 * 4 + 2 * (lane >> 4) + p;
        const int co   = nt * 16 + (lane & 15);
        ldsw[i] = wmain[(co * CC + c) * K2C + k];
    }
    __syncthreads();

    const int lane   = threadIdx.x & 31;
    const int waveId = threadIdx.x >> 5;
    const int tile   = blockIdx.x * 8 + waveId;      // 8192 tiles total
    const int b      = tile / (HW / 16);
    const int rem    = tile % (HW / 16);
    const int h      = rem / (WW / 16);
    const int w0     = (rem % (WW / 16)) * 16;
    const int pos    = lane & 15;
    const int wx     = w0 + pos;
    const int kpair  = lane >> 4;

    v8f acc0 = {}, acc1 = {}, acc2 = {}, acc3 = {};
    const float* xb   = xin + (size_t)b * CC * HW;
    const float* offb = offA + (size_t)b * NOFF * HW + h * WW + wx;

    for (int k = 0; k < K2C; ++k) {
        // Per-lane bilinear sampling parameters (lanes L and L+16 duplicate)
        const float dy = offb[(2 * k) * HW];
        const float dx = offb[(2 * k + 1) * HW];
        const float m  = 1.0f / (1.0f + __expf(-offb[(18 + k) * HW]));
        const int  ky  = k / 3 - 1;
        const int  kx  = k % 3 - 1;
        const float py = dy + (float)(h + ky);
        const float px = dx + (float)(wx + kx);
        const float y0f = floorf(py), x0f = floorf(px);
        const float ly = py - y0f,   lx = px - x0f;
        const int y0 = (int)y0f, x0 = (int)x0f;
        const int y1 = y0 + 1,   x1 = x0 + 1;
        const float vy0 = ((y0 >= 0) & (y0 < HH)) ? 1.f : 0.f;
        const float vy1 = ((y1 >= 0) & (y1 < HH)) ? 1.f : 0.f;
        const float vx0 = ((x0 >= 0) & (x0 < WW)) ? 1.f : 0.f;
        const float vx1 = ((x1 >= 0) & (x1 < WW)) ? 1.f : 0.f;
        const int cy0 = min(max(y0, 0), HH - 1), cy1 = min(max(y1, 0), HH - 1);
        const int cx0 = min(max(x0, 0), WW - 1), cx1 = min(max(x1, 0), WW - 1);
        const int i00 = cy0 * WW + cx0, i01 = cy0 * WW + cx1;
        const int i10 = cy1 * WW + cx0, i11 = cy1 * WW + cx1;
        const float w00 = (1.f - ly) * (1.f - lx) * vy0 * vx0 * m;
        const float w01 = (1.f - ly) * lx         * vy0 * vx1 * m;
        const float w10 = ly         * (1.f - lx) * vy1 * vx0 * m;
        const float w11 = ly         * lx         * vy1 * vx1 * m;

        const float* pc  = xb + (2 * kpair) * HW;
        const v2f*   ldk = (const v2f*)(ldsw + k * (16 * 4 * 32 * 2));

        #pragma unroll 4
        for (int c4 = 0; c4 < 16; ++c4) {
            const float* p0 = pc + c4 * 4 * HW;
            const float* p1 = p0 + HW;
            // A fragment: bilinear gather for 2 channels; the 4 corner
            // addresses/weights are shared across all 64 output channels.
            const float a0 = w00 * p0[i00] + w01 * p0[i01] + w10 * p0[i10] + w11 * p0[i11];
            const float a1 = w00 * p1[i00] + w01 * p1[i01] + w10 * p1[i10] + w11 * p1[i11];
            v2f A; A.x = a0; A.y = a1;

            const v2f* bp = ldk + c4 * (4 * 32);
            const v2f B0 = bp[0 * 32 + lane];
            const v2f B1 = bp[1 * 32 + lane];
            const v2f B2 = bp[2 * 32 + lane];
            const v2f B3 = bp[3 * 32 + lane];

            acc0 = __builtin_amdgcn_wmma_f32_16x16x4_f32(false, A, false, B0, (short)0, acc0, false, false);
            acc1 = __builtin_amdgcn_wmma_f32_16x16x4_f32(false, A, false, B1, (short)0, acc1, false, false);
            acc2 = __builtin_amdgcn_wmma_f32_16x16x4_f32(false, A, false, B2, (short)0, acc2, false, false);
            acc3 = __builtin_amdgcn_wmma_f32_16x16x4_f32(false, A, false, B3, (short)0, acc3, false, false);
        }
    }

    // Epilogue: D layout -> vgpr r, lane: M = r + 8*(lane>>4), N = lane&15
    v8f accs[4] = { acc0, acc1, acc2, acc3 };
    const int mbase = 8 * (lane >> 4);
    float* outb = out + (size_t)b * CO_ * HW + h * WW + w0;
    #pragma unroll
    for (int nt = 0; nt < 4; ++nt) {
        const int co   = nt * 16 + (lane & 15);
        const float bias = bconv[co];
        const float inv  = gamma[co] * rsqrtf(rvar[co] + BN_EPS);
        const float bet  = beta[co] - rmean[co] * inv;
        float* po = outb + co * HW;
        #pragma unroll
        for (int r = 0; r < 8; ++r) {
            float v = accs[nt][r] + bias;
            v = fmaf(v, inv, bet);
            po[mbase + r] = fmaxf(v, 0.0f);
        }
    }
}

// ---------------------------------------------------------------------------
// Launcher
// ---------------------------------------------------------------------------
extern "C" void kernel_launch(void* const* d_in, const int* in_sizes, int n_in,
                              void* d_out, int out_size, void* d_ws, size_t ws_size,
                              hipStream_t stream)
{
    const float* x     = (const float*)d_in[0];
    const float* w_off = (const float*)d_in[1];
    const float* b_off = (const float*)d_in[2];
    const float* w     = (const float*)d_in[3];
    const float* bconv = (const float*)d_in[4];
    const float* gamma = (const float*)d_in[5];
    const float* beta  = (const float*)d_in[6];
    const float* rmean = (const float*)d_in[7];
    const float* rvar  = (const float*)d_in[8];
    float* out = (float*)d_out;
    float* offA = (float*)d_ws;                 // [B][27][H][W] = 14.2 MB

    // Kernel 1: WMMA offset conv. 8192 M-tiles, 8 waves/block -> 1024 blocks,
    // 72 KB dynamic LDS (4 blocks/WGP under the 320 KB budget).
    offset_kernel<<<1024, 256, 73728, stream>>>(x, w_off, b_off, offA);

    // Kernel 2: WMMA deform GEMM. 8192 M-tiles, 8 waves/block -> 1024 blocks,
    // 144 KB dynamic LDS (2 blocks/WGP -> 4 waves/SIMD).
    deform_gemm_kernel<<<1024, 256, 147456, stream>>>(x, w, bconv, gamma, beta,
                                                      rmean, rvar, offA, out);
}